// SantaFeSNN_44152263803036
// MI455X (gfx1250) — compile-verified
//
#include <hip/hip_runtime.h>
#include <stdint.h>

typedef __attribute__((ext_vector_type(2))) float v2f;
typedef __attribute__((ext_vector_type(8))) float v8f;

// Problem dims (fixed by reference)
constexpr int BATCH = 256, SEQ = 128, IN = 512, HID = 2048, ACT = 512;
// Tiling
constexpr int NT = 128;          // N tile (hidden / action columns per WG)
constexpr int KC = 32;           // K chunk staged in LDS
constexpr int AS_STRIDE = KC + 1;   // 33  (pad to avoid bank conflicts)
constexpr int BS_STRIDE = NT + 1;   // 129
constexpr int SS = NT + 1;          // 129 (scan buffer stride)

constexpr float BETA = 0.9f;
constexpr float THR  = 1.0f;

// ---------------------------------------------------------------------------
// Kernel 1: fused  cur = x @ W1^T + b1  (fp32 WMMA)  ->  leaky scan  ->
// bit-packed spikes.  One WG = one batch index b (M=128=all seq) x 128 hidden.
// ---------------------------------------------------------------------------
__global__ void __launch_bounds__(256)
snn_fc1_scan(const float* __restrict__ x, const float* __restrict__ W1,
             const float* __restrict__ b1, uint32_t* __restrict__ spkC)
{
    // Overlay: during GEMM, smem holds AS (128x33) + BS (32x129) = 8352 floats.
    // After GEMM, the full buffer becomes the scan matrix S[128 s][129].
    __shared__ float smem[SEQ * SS];          // 16512 floats = 66 KB
    float* AS = smem;                          // [128][33]
    float* BS = smem + SEQ * AS_STRIDE;        // [32][129]

    const int tid  = threadIdx.x;
    const int lane = tid & 31;
    const int wave = tid >> 5;        // 0..7
    const int wm   = wave & 1;        // M half: rows wm*64 + ms*16
    const int wn   = wave >> 1;       // N quarter: cols wn*32 + ns*16
    const int b     = blockIdx.y;
    const int hBase = blockIdx.x * NT;

    v8f acc[4][2] = {};               // 4 M-subtiles x 2 N-subtiles

    const int koff = (lane >> 4) * 2; // lanes 16-31 hold K+2,K+3 (ISA A/B layout)
    const int m15  = lane & 15;

    for (int kc = 0; kc < IN; kc += KC) {
        // --- stage A: x[b, s, kc..kc+31]  -> AS[s][k]
        for (int j = tid; j < SEQ * KC / 4; j += 256) {
            const int s = j >> 3, q = j & 7;
            const float4 v = *reinterpret_cast<const float4*>(
                x + (size_t)(b * SEQ + s) * IN + kc + q * 4);
            float* dst = AS + s * AS_STRIDE + q * 4;
            dst[0] = v.x; dst[1] = v.y; dst[2] = v.z; dst[3] = v.w;
        }
        // --- stage B (transposed): W1[hBase+h, kc+i] -> BS[i][h]
        for (int j = tid; j < NT * KC / 4; j += 256) {
            const int h = j >> 3, q = j & 7;
            const float4 v = *reinterpret_cast<const float4*>(
                W1 + (size_t)(hBase + h) * IN + kc + q * 4);
            BS[(q * 4 + 0) * BS_STRIDE + h] = v.x;
            BS[(q * 4 + 1) * BS_STRIDE + h] = v.y;
            BS[(q * 4 + 2) * BS_STRIDE + h] = v.z;
            BS[(q * 4 + 3) * BS_STRIDE + h] = v.w;
        }
        __syncthreads();

        for (int k = 0; k < KC; k += 4) {
            v2f afr[4], bfr[2];
            for (int ms = 0; ms < 4; ms++) {
                const float* ap = AS + (wm * 64 + ms * 16 + m15) * AS_STRIDE + k + koff;
                afr[ms][0] = ap[0];
                afr[ms][1] = ap[1];
            }
            for (int ns = 0; ns < 2; ns++) {
                const float* bp = BS + (k + koff) * BS_STRIDE + wn * 32 + ns * 16 + m15;
                bfr[ns][0] = bp[0];
                bfr[ns][1] = bp[BS_STRIDE];
            }
            for (int ms = 0; ms < 4; ms++)
                for (int ns = 0; ns < 2; ns++)
                    acc[ms][ns] = __builtin_amdgcn_wmma_f32_16x16x4_f32(
                        false, afr[ms], false, bfr[ns],
                        (short)0, acc[ms][ns], false, false);
        }
        __syncthreads();
    }

    // --- epilogue: cur + b1 -> scan buffer S[s][h]
    {
        const int mSub = wm * 64 + ((lane >> 4) * 8);  // D layout: M = r + 8*(lane>=16)
        float bias[2];
        bias[0] = b1[hBase + wn * 32 + m15];
        bias[1] = b1[hBase + wn * 32 + 16 + m15];
        for (int ms = 0; ms < 4; ms++)
            for (int ns = 0; ns < 2; ns++)
                for (int r = 0; r < 8; r++) {
                    const int s = mSub + ms * 16 + r;
                    const int h = wn * 32 + ns * 16 + m15;
                    smem[s * SS + h] = acc[ms][ns][r] + bias[ns];
                }
    }
    __syncthreads();

    // --- leaky scan: thread t owns hidden column t; pack 128 spike bits.
    if (tid < NT) {
        const int col = tid;
        float mem = 0.0f;
        uint32_t bits[4];
        uint32_t w = 0;
        for (int s = 0; s < SEQ; s++) {
            const float c = smem[s * SS + col];
            const float reset = (mem > THR) ? THR : 0.0f;  // from previous mem
            mem = BETA * mem + c - reset;
            if (mem > THR) w |= (1u << (s & 31));          // spike = (mem - 1 > 0)
            if ((s & 31) == 31) { bits[s >> 5] = w; w = 0; }
        }
        uint4 o; o.x = bits[0]; o.y = bits[1]; o.z = bits[2]; o.w = bits[3];
        *reinterpret_cast<uint4*>(spkC + ((size_t)b * HID + hBase + col) * 4) = o;
    }
}

// ---------------------------------------------------------------------------
// Kernel 2: out = spk @ W2^T + b2 (fp32 WMMA), spikes unpacked from bits.
// One WG = batch b (M=128 rows = all seq) x 128 actions.
// ---------------------------------------------------------------------------
__global__ void __launch_bounds__(256)
snn_fc2(const uint32_t* __restrict__ spkC, const float* __restrict__ W2,
        const float* __restrict__ b2, float* __restrict__ out)
{
    __shared__ float AS[SEQ * AS_STRIDE];   // [128 s][32 k]
    __shared__ float BS[KC * BS_STRIDE];    // [32 k][128 a]

    const int tid  = threadIdx.x;
    const int lane = tid & 31;
    const int wave = tid >> 5;
    const int wm   = wave & 1;
    const int wn   = wave >> 1;
    const int b     = blockIdx.y;
    const int aBase = blockIdx.x * NT;

    v8f acc[4][2] = {};
    const int koff = (lane >> 4) * 2;
    const int m15  = lane & 15;

    for (int kc = 0; kc < HID; kc += KC) {
        // --- stage A from bit-packed spikes: AS[s][h'] for h' in [0,32)
        if (tid < 128) {
            const int hh = tid >> 2, sw = tid & 3;
            const uint32_t wbits = spkC[((size_t)b * HID + kc + hh) * 4 + sw];
            for (int i = 0; i < 32; i++)
                AS[(sw * 32 + i) * AS_STRIDE + hh] = (wbits >> i) & 1u ? 1.0f : 0.0f;
        }
        // --- stage B (transposed): W2[aBase+a, kc+i] -> BS[i][a]
        for (int j = tid; j < NT * KC / 4; j += 256) {
            const int a = j >> 3, q = j & 7;
            const float4 v = *reinterpret_cast<const float4*>(
                W2 + (size_t)(aBase + a) * HID + kc + q * 4);
            BS[(q * 4 + 0) * BS_STRIDE + a] = v.x;
            BS[(q * 4 + 1) * BS_STRIDE + a] = v.y;
            BS[(q * 4 + 2) * BS_STRIDE + a] = v.z;
            BS[(q * 4 + 3) * BS_STRIDE + a] = v.w;
        }
        __syncthreads();

        for (int k = 0; k < KC; k += 4) {
            v2f afr[4], bfr[2];
            for (int ms = 0; ms < 4; ms++) {
                const float* ap = AS + (wm * 64 + ms * 16 + m15) * AS_STRIDE + k + koff;
                afr[ms][0] = ap[0];
                afr[ms][1] = ap[1];
            }
            for (int ns = 0; ns < 2; ns++) {
                const float* bp = BS + (k + koff) * BS_STRIDE + wn * 32 + ns * 16 + m15;
                bfr[ns][0] = bp[0];
                bfr[ns][1] = bp[BS_STRIDE];
            }
            for (int ms = 0; ms < 4; ms++)
                for (int ns = 0; ns < 2; ns++)
                    acc[ms][ns] = __builtin_amdgcn_wmma_f32_16x16x4_f32(
                        false, afr[ms], false, bfr[ns],
                        (short)0, acc[ms][ns], false, false);
        }
        __syncthreads();
    }

    // --- epilogue: + b2, write out[(b*SEQ+s)*ACT + a]
    const int mSub = wm * 64 + ((lane >> 4) * 8);
    float bias[2];
    bias[0] = b2[aBase + wn * 32 + m15];
    bias[1] = b2[aBase + wn * 32 + 16 + m15];
    for (int ms = 0; ms < 4; ms++)
        for (int ns = 0; ns < 2; ns++)
            for (int r = 0; r < 8; r++) {
                const int s = mSub + ms * 16 + r;
                const int a = aBase + wn * 32 + ns * 16 + m15;
                out[(size_t)(b * SEQ + s) * ACT + a] = acc[ms][ns][r] + bias[ns];
            }
}

// ---------------------------------------------------------------------------
extern "C" void kernel_launch(void* const* d_in, const int* in_sizes, int n_in,
                              void* d_out, int out_size, void* d_ws, size_t ws_size,
                              hipStream_t stream) {
    const float* x  = (const float*)d_in[0];
    const float* W1 = (const float*)d_in[1];
    const float* b1 = (const float*)d_in[2];
    const float* W2 = (const float*)d_in[3];
    const float* b2 = (const float*)d_in[4];
    float* out = (float*)d_out;
    uint32_t* spkC = (uint32_t*)d_ws;   // 256*2048*4 words = 8 MB bit-packed spikes

    dim3 g1(HID / NT, BATCH);   // (16, 256)
    snn_fc1_scan<<<g1, dim3(256), 0, stream>>>(x, W1, b1, spkC);

    dim3 g2(ACT / NT, BATCH);   // (4, 256)
    snn_fc2<<<g2, dim3(256), 0, stream>>>(spkC, W2, b2, out);
}